// scale_layer_2439541424302
// MI455X (gfx1250) — compile-verified
//
#include <hip/hip_runtime.h>

// Problem constants (from the reference)
#define BATCH 32
#define H     512
#define W     512
#define SCALE 4
#define PAD   128
#define GY    (H + 2 * PAD)       // 768 padded rows  (pre-upsample)
#define GX    (W + 2 * PAD)       // 768 padded cols  (pre-upsample)
#define OH    (GY * SCALE)        // 3072 output rows
#define OW    (GX * SCALE)        // 3072 output cols

typedef float float4v __attribute__((ext_vector_type(4)));

// One thread produces one 4x4 output tile from one (padded) input pixel.
//   - 1 coalesced global_load_b32 (skipped in the zero-pad region)
//   - 4 non-temporal global_store_b128 (one 16B store per replicated row)
// Pure store-bandwidth-bound: ~1.21 GB written, ~33.5 MB read.
__global__ __launch_bounds__(256) void upsample_pad_nn_kernel(
    const float* __restrict__ in, float* __restrict__ out)
{
    const int gxi = blockIdx.x * 256 + threadIdx.x;  // padded col index, 0..GX-1
    const int gyi = blockIdx.y;                      // padded row index, 0..GY-1
    const int b   = blockIdx.z;                      // batch index

    // Map padded coordinate -> source coordinate; outside -> zero.
    const int ix = gxi - PAD;
    const int iy = gyi - PAD;

    float v = 0.0f;
    if (((unsigned)ix < (unsigned)W) & ((unsigned)iy < (unsigned)H)) {
        // Streamed once; non-temporal so the 33.5MB input doesn't linger.
        v = __builtin_nontemporal_load(
                &in[((size_t)b * H + (size_t)iy) * W + (size_t)ix]);
    }

    const float4v vec = {v, v, v, v};

    // Output base of this thread's 4x4 tile. gxi*SCALE is a multiple of 4
    // floats -> 16-byte aligned -> global_store_b128.
    size_t base = ((size_t)b * OH + (size_t)gyi * SCALE) * OW
                + (size_t)gxi * SCALE;

#pragma unroll
    for (int r = 0; r < SCALE; ++r) {
        // NT store: 1.2GB output >> 192MB L2, written once, never re-read.
        __builtin_nontemporal_store(
            vec, (float4v*)(out + base + (size_t)r * OW));
    }
}

extern "C" void kernel_launch(void* const* d_in, const int* in_sizes, int n_in,
                              void* d_out, int out_size, void* d_ws, size_t ws_size,
                              hipStream_t stream)
{
    (void)in_sizes; (void)n_in; (void)out_size; (void)d_ws; (void)ws_size;

    const float* x = (const float*)d_in[0];
    float* out     = (float*)d_out;

    // GX = 768 = 3 * 256 -> exact cover, no tail handling needed.
    dim3 grid(GX / 256, GY, BATCH);
    dim3 block(256, 1, 1);
    upsample_pad_nn_kernel<<<grid, block, 0, stream>>>(x, out);
}